// SparseFlowModel_79096117723798
// MI455X (gfx1250) — compile-verified
//
#include <hip/hip_runtime.h>
#include <stdint.h>

#define BATCH 4
#define NBLK 2048
#define COORDS_PER_BATCH (NBLK * 3)
#define HALF_COORDS (COORDS_PER_BATCH / 2)  // 3072 dwords = 12 KB per descriptor
#define HASH_SIZE 8192
#define HASH_MASK (HASH_SIZE - 1)
#define EMPTY_KEY ((int)0x80000000)
#define P1 100003
#define P2 1009
#define DINV (1.0f / 15.0f)

typedef unsigned int u32;
typedef __attribute__((ext_vector_type(4))) u32 u32x4;
typedef __attribute__((ext_vector_type(8))) int i32x8;
typedef __attribute__((ext_vector_type(4))) int i32x4;

#if __has_builtin(__builtin_amdgcn_tensor_load_to_lds) && \
    __has_builtin(__builtin_amdgcn_s_wait_tensorcnt)
#define USE_TDM 1
#warning "CDNA5 probe: tensor_load_to_lds + s_wait_tensorcnt builtins AVAILABLE (TDM path compiled in)"
#else
#warning "CDNA5 probe: TDM builtins NOT available (plain-load fallback compiled in)"
#endif

#if __has_builtin(__builtin_amdgcn_global_load_async_to_lds_b32)
#warning "CDNA5 probe: __builtin_amdgcn_global_load_async_to_lds_b32 AVAILABLE"
#endif
#if __has_builtin(__builtin_amdgcn_s_wait_asynccnt)
#warning "CDNA5 probe: __builtin_amdgcn_s_wait_asynccnt AVAILABLE"
#endif

// First 6 entries == FACE_OFFSETS (order matters: pattern bit f == face f).
__device__ const int OFF[26][3] = {
    {-1,0,0},{1,0,0},{0,-1,0},{0,1,0},{0,0,-1},{0,0,1},
    {-1,-1,0},{-1,1,0},{1,-1,0},{1,1,0},
    {-1,0,-1},{-1,0,1},{1,0,-1},{1,0,1},
    {0,-1,-1},{0,-1,1},{0,1,-1},{0,1,1},
    {-1,-1,-1},{-1,-1,1},{-1,1,-1},{-1,1,1},
    {1,-1,-1},{1,-1,1},{1,1,-1},{1,1,1}};

__device__ __forceinline__ u32 hash_key(int key) {
  return ((u32)key * 2654435761u) >> 19;  // -> [0, 8191]
}

#ifdef USE_TDM
// Issue one TDM descriptor: contiguous copy of `ndw` dwords global -> LDS.
// 1 row x ndw elements, data_size=4B (ISA 8.3/8.4 bit layout).
__device__ __forceinline__ void tdm_copy_dwords(const int* gsrc, int* ldst, int ndw) {
  uint64_t ga = (uint64_t)(uintptr_t)gsrc;
  u32 la = (u32)(uintptr_t)ldst;  // low 32 bits of generic ptr = LDS byte addr
  u32x4 g0;
  g0[0] = 1u;                                             // count=1, user desc
  g0[1] = la;                                             // lds_addr
  g0[2] = (u32)(ga & 0xFFFFFFFFu);                        // global_addr[31:0]
  g0[3] = (u32)((ga >> 32) & 0x01FFFFFFu) | 0x80000000u;  // addr hi | type=2
  i32x8 g1;
  g1[0] = 0x20000;                       // data_size=2 (4-byte elements)
  g1[1] = (int)((u32)ndw << 16);         // tensor_dim0[15:0]  (bits 63:48)
  g1[2] = (int)(((u32)ndw >> 16) | (1u << 16));  // dim0 hi | tensor_dim1=1
  g1[3] = (int)((u32)ndw << 16);         // tile_dim0 = ndw    (bits 127:112)
  g1[4] = 1;                             // tile_dim1 = 1
  g1[5] = ndw;                           // tensor_dim0_stride lo32
  g1[6] = 0;
  g1[7] = 0;
  i32x4 gz = {0, 0, 0, 0};
#if __clang_major__ >= 23
  i32x8 gz8 = {0, 0, 0, 0, 0, 0, 0, 0};
  __builtin_amdgcn_tensor_load_to_lds(g0, g1, gz, gz, gz8, 0);
#else
  __builtin_amdgcn_tensor_load_to_lds(g0, g1, gz, gz, 0);
#endif
}
#endif

// Kernel 1: one workgroup per batch. Stage coords to LDS (TDM if available),
// build an LDS open-addressing hash set of the 2048 keys, then do the 26
// membership probes per block: write occ floats and the 6-bit face pattern.
__global__ __launch_bounds__(256) void occ_kernel(const int* __restrict__ coords,
                                                  float* __restrict__ occ,
                                                  int* __restrict__ patterns) {
  __shared__ int table[HASH_SIZE];
  __shared__ int scoords[COORDS_PER_BATCH];
  const int b = blockIdx.x;
  const int tid = threadIdx.x;
  const int* cbase = coords + (size_t)b * COORDS_PER_BATCH;

  for (int i = tid; i < HASH_SIZE; i += 256) table[i] = EMPTY_KEY;

#ifdef USE_TDM
  // Waves 0 and 1 each DMA half the coords (12 KB) and wait their TENSORcnt.
  if (tid == 0 || tid == 32) {
    const int part = (tid == 0) ? 0 : 1;
    tdm_copy_dwords(cbase + part * HALF_COORDS, &scoords[part * HALF_COORDS],
                    HALF_COORDS);
    __builtin_amdgcn_s_wait_tensorcnt(0);
  }
#else
  __builtin_prefetch(cbase + tid * 24, 0, 0);  // global_prefetch_b8
  for (int i = tid; i < COORDS_PER_BATCH; i += 256) scoords[i] = cbase[i];
#endif
  __syncthreads();

  // Insert the batch's keys (ds_cmpstore_rtn_b32 via atomicCAS).
  for (int n = tid; n < NBLK; n += 256) {
    int key = scoords[n * 3] * P1 + scoords[n * 3 + 1] * P2 + scoords[n * 3 + 2];
    u32 h = hash_key(key);
    for (;;) {
      int prev = atomicCAS(&table[h], EMPTY_KEY, key);
      if (prev == EMPTY_KEY || prev == key) break;
      h = (h + 1) & HASH_MASK;
    }
  }
  __syncthreads();

  // 26 membership probes per block.
  for (int n = tid; n < NBLK; n += 256) {
    const int x = scoords[n * 3], y = scoords[n * 3 + 1], z = scoords[n * 3 + 2];
    float* orow = occ + ((size_t)b * NBLK + n) * 26;
    int pat = 0;
#pragma unroll
    for (int o = 0; o < 26; ++o) {
      int key = (x + OFF[o][0]) * P1 + (y + OFF[o][1]) * P2 + (z + OFF[o][2]);
      u32 h = hash_key(key);
      bool found = false;
      for (;;) {
        int v = table[h];
        if (v == key) { found = true; break; }
        if (v == EMPTY_KEY) break;
        h = (h + 1) & HASH_MASK;
      }
      orow[o] = found ? 1.0f : 0.0f;
      if (o < 6) pat |= found ? (1 << o) : 0;
    }
    patterns[b * NBLK + n] = pat;
  }
}

// Kernel 2: the bandwidth kernel. One WG per (b,n) block; each thread owns one
// (i,j) column of 16 voxels, writes 4 x float4 (global_store_b128). 16 KB/WG.
__global__ __launch_bounds__(256) void prior_kernel(const int* __restrict__ patterns,
                                                    float* __restrict__ prior) {
  const int blk = blockIdx.x;            // b*NBLK + n
  const int tid = threadIdx.x;           // i = tid>>4, j = tid&15
  const int pat = patterns[blk];

  const float fi = (float)(tid >> 4) * DINV;
  const float fj = (float)(tid & 15) * DINV;
  const float a = (pat & 1) ? 1.0f : fi;
  const float b = (pat & 2) ? 1.0f : 1.0f - fi;
  const float c = (pat & 4) ? 1.0f : fj;
  const float d = (pat & 8) ? 1.0f : 1.0f - fj;
  const float base = fminf(fminf(a, b), fminf(c, d));
  const bool h4 = (pat & 16) != 0;
  const bool h5 = (pat & 32) != 0;

  float4* out = (float4*)(prior + (size_t)blk * 4096 + (size_t)tid * 16);
#pragma unroll
  for (int q = 0; q < 4; ++q) {
    float4 v;
    float* vp = (float*)&v;
#pragma unroll
    for (int r = 0; r < 4; ++r) {
      const int k = q * 4 + r;
      const float e = h4 ? 1.0f : (float)k * DINV;
      const float f = h5 ? 1.0f : 1.0f - (float)k * DINV;
      vp[r] = fminf(base, fminf(e, f));
    }
    out[q] = v;
  }
}

extern "C" void kernel_launch(void* const* d_in, const int* in_sizes, int n_in,
                              void* d_out, int out_size, void* d_ws, size_t ws_size,
                              hipStream_t stream) {
  const int* coords = (const int*)d_in[0];
  float* prior = (float*)d_out;
  float* occ = (float*)d_out + (size_t)BATCH * NBLK * 4096;  // outputs concat flat
  int* patterns = (int*)d_ws;  // BATCH*NBLK ints = 32 KB, rewritten every call

  occ_kernel<<<BATCH, 256, 0, stream>>>(coords, occ, patterns);
  prior_kernel<<<BATCH * NBLK, 256, 0, stream>>>(patterns, prior);
}